// MPNNs_curv_24266565222960
// MI455X (gfx1250) — compile-verified
//
#include <hip/hip_runtime.h>

typedef float v2f __attribute__((ext_vector_type(2)));
typedef float v8f __attribute__((ext_vector_type(8)));

#define DD 256
#define EPSV 1e-10f
#define LNEPS 1e-5f

// ---------------- degree / normalization ----------------
__global__ void k_fill1(float* p, int n) {
    int i = blockIdx.x * blockDim.x + threadIdx.x;
    if (i < n) p[i] = 1.0f;  // self-loop contributes 1
}

__global__ void k_deg_count(const int* __restrict__ col, float* deg, int E) {
    int e = blockIdx.x * blockDim.x + threadIdx.x;
    if (e < E) atomicAdd(&deg[col[e]], 1.0f);
}

__global__ void k_rsqrt_inplace(float* p, int n) {
    int i = blockIdx.x * blockDim.x + threadIdx.x;
    if (i < n) p[i] = rsqrtf(p[i]);
}

// ---------------- fourier z matrices (2 x 256 x 256) ----------------
__global__ void k_compute_z(const float* __restrict__ feat, const float* __restrict__ wf,
                            const float* __restrict__ bias_f, const float* __restrict__ lam,
                            float* __restrict__ z) {
    int idx = blockIdx.x * blockDim.x + threadIdx.x;   // [0, 2*256*256)
    int c  = idx >> 16;
    int ij = idx & 0xFFFF;
    int i  = ij >> 8;
    int j  = ij & 255;
    const float* xf = feat + (c << 16) + i * DD;
    const float* w  = wf   + (c << 16) + j * DD;
    float div = 0.0f, xx = 0.0f;
    for (int k = 0; k < DD; ++k) {
        float a = xf[k];
        float d = a - w[k];
        div += d * d;
        xx  += a * a;
    }
    float curv = (c == 0) ? 1.0f : -1.0f;
    float num  = 1.0f + curv * xx;
    float dist = logf(num / (div + EPSV) + EPSV);
    z[idx] = expf(0.5f * (DD - 1) * dist) * cosf(lam[idx] * dist + bias_f[idx]);
}

// ---------------- pack W into WMMA-fragment layout ----------------
// Wp[(k/4)*1024 + col*4 + (k%4)] = W[k*256 + col]
// -> a B fragment (two consecutive K rows for one col) is one aligned 8B load.
__global__ void k_pack(const float* __restrict__ src, float* __restrict__ dst, int nmat) {
    int idx = blockIdx.x * blockDim.x + threadIdx.x;
    if (idx >= nmat * DD * DD) return;
    int mat = idx >> 16;
    int rc  = idx & 0xFFFF;
    int r   = rc >> 8;    // k row
    int c   = rc & 255;   // col
    dst[(mat << 16) + (r >> 2) * (DD * 4) + c * 4 + (r & 3)] = src[idx];
}

// ---------------- WMMA f32 GEMM: Out[M,256] = A[M,256] @ W[256,256] (+Cin)(+bias) ----------------
// W is pre-packed (k_pack layout). block = 128 threads (4 waves).
// Block owns 16 rows; wave w owns cols [w*64, w*64+64).
template <bool ADD_C, bool ADD_BIAS>
__global__ __launch_bounds__(128) void k_gemm(const float* __restrict__ A,
                                              const float* __restrict__ Wp,
                                              const float* Cin, const float* __restrict__ bias,
                                              float* Out) {
    __shared__ float As[16 * DD];   // 16KB A tile
    const int tid = threadIdx.x;
    const int m0  = blockIdx.x * 16;

    // cooperative, coalesced A-tile load (4096 contiguous floats)
    {
        const float4* src = (const float4*)(A + (size_t)m0 * DD);
        float4* dst = (float4*)As;
        #pragma unroll
        for (int i = 0; i < (16 * DD / 4) / 128; ++i)
            dst[tid + i * 128] = src[tid + i * 128];
    }
    __syncthreads();

    const int lane  = tid & 31;
    const int nbase = (tid >> 5) * 64;
    const int al    = lane & 15;
    const int khalf = (lane >= 16) ? 2 : 0;   // A/B frag: lanes 16..31 carry K+2,K+3

    v8f acc0 = {}, acc1 = {}, acc2 = {}, acc3 = {};

    // per-lane base into packed W: fragment = 8B at (kgroup*1024 + col*4 + khalf)
    const float* wbase = Wp + (nbase + al) * 4 + khalf;

    #pragma unroll 4
    for (int k = 0; k < DD; k += 4) {
        const int krow = k + khalf;
        // A fragment: v0 = A[m, krow], v1 = A[m, krow+1] (krow even -> 8B aligned)
        v2f a = *(const v2f*)(&As[al * DD + krow]);
        // B fragments: packed -> one b64 load each; 16 lanes span 256B contiguous
        const float* wp = wbase + (k >> 2) * (DD * 4);
        v2f b0 = *(const v2f*)(wp);
        v2f b1 = *(const v2f*)(wp + 64);
        v2f b2 = *(const v2f*)(wp + 128);
        v2f b3 = *(const v2f*)(wp + 192);
        __builtin_prefetch(wp + DD * 4, 0, 3);  // next k-group (global_prefetch_b8)
        acc0 = __builtin_amdgcn_wmma_f32_16x16x4_f32(false, a, false, b0, (short)0, acc0, false, false);
        acc1 = __builtin_amdgcn_wmma_f32_16x16x4_f32(false, a, false, b1, (short)0, acc1, false, false);
        acc2 = __builtin_amdgcn_wmma_f32_16x16x4_f32(false, a, false, b2, (short)0, acc2, false, false);
        acc3 = __builtin_amdgcn_wmma_f32_16x16x4_f32(false, a, false, b3, (short)0, acc3, false, false);
    }

    // epilogue: C/D layout — VGPR r: lanes 0-15 -> row m0+r, lanes 16-31 -> row m0+8+r
    const int rowoff = (lane >= 16) ? 8 : 0;
    #pragma unroll
    for (int r = 0; r < 8; ++r) {
        const int row = m0 + r + rowoff;
        float* orow = Out + (size_t)row * DD + nbase + al;
        float v0 = acc0[r], v1 = acc1[r], v2 = acc2[r], v3 = acc3[r];
        if (ADD_C) {
            const float* crow = Cin + (size_t)row * DD + nbase + al;
            v0 += crow[0]; v1 += crow[16]; v2 += crow[32]; v3 += crow[48];
        }
        if (ADD_BIAS) {
            const float* bb = bias + nbase + al;
            v0 += bb[0]; v1 += bb[16]; v2 += bb[32]; v3 += bb[48];
        }
        orow[0] = v0; orow[16] = v1; orow[32] = v2; orow[48] = v3;
    }
}

// ---------------- agg init: self-loop term + conv bias ----------------
__global__ void k_agg_init(const float* __restrict__ Hw, const float* __restrict__ dinv,
                           const float* __restrict__ bias, float* __restrict__ agg, int M) {
    int i = blockIdx.x * blockDim.x + threadIdx.x;  // over M * 64 float4s
    if (i >= M * (DD / 4)) return;
    int nrow = i / (DD / 4);
    int dc   = (i % (DD / 4)) * 4;
    float s = dinv[nrow]; s = s * s;
    float4 h = ((const float4*)Hw)[i];
    float4 o;
    o.x = h.x * s + bias[dc + 0];
    o.y = h.y * s + bias[dc + 1];
    o.z = h.z * s + bias[dc + 2];
    o.w = h.w * s + bias[dc + 3];
    ((float4*)agg)[i] = o;
}

// ---------------- edge scatter: agg[col] += Hw[row] * dinv[row]*dinv[col] ----------------
__global__ void k_scatter(const float* __restrict__ Hw, const int* __restrict__ row,
                          const int* __restrict__ col, const float* __restrict__ dinv,
                          float* agg) {
    const int e = blockIdx.x;
    const int d = threadIdx.x;
    const int r = row[e];
    const int c = col[e];
    const float w = dinv[r] * dinv[c];
    atomicAdd(&agg[(size_t)c * DD + d], Hw[(size_t)r * DD + d] * w);
}

// ---------------- layernorm (+relu), one wave32 per row ----------------
template <bool RELU>
__global__ void k_ln(const float* __restrict__ X, const float* __restrict__ g,
                     const float* __restrict__ b, float* __restrict__ Y, int M) {
    const int wid  = threadIdx.x >> 5;
    const int lane = threadIdx.x & 31;
    const int row  = blockIdx.x * 8 + wid;
    if (row >= M) return;
    const float* x = X + (size_t)row * DD;
    float v[8];
    float s = 0.0f, sq = 0.0f;
    #pragma unroll
    for (int j = 0; j < 8; ++j) {
        float t = x[lane + j * 32];
        v[j] = t; s += t; sq += t * t;
    }
    #pragma unroll
    for (int m = 16; m >= 1; m >>= 1) {
        s  += __shfl_xor(s,  m, 32);
        sq += __shfl_xor(sq, m, 32);
    }
    const float mean = s * (1.0f / DD);
    const float var  = sq * (1.0f / DD) - mean * mean;
    const float rstd = rsqrtf(var + LNEPS);
    float* y = Y + (size_t)row * DD;
    #pragma unroll
    for (int j = 0; j < 8; ++j) {
        const int dc = lane + j * 32;
        float o = (v[j] - mean) * rstd * g[dc] + b[dc];
        if (RELU) o = fmaxf(o, 0.0f);
        y[dc] = o;
    }
}

extern "C" void kernel_launch(void* const* d_in, const int* in_sizes, int n_in,
                              void* d_out, int out_size, void* d_ws, size_t ws_size,
                              hipStream_t stream) {
    const float* x      = (const float*)d_in[0];
    const float* conv_W = (const float*)d_in[1];
    const float* conv_b = (const float*)d_in[2];
    const float* lin_W  = (const float*)d_in[3];
    const float* lin_b  = (const float*)d_in[4];
    const float* ln_g   = (const float*)d_in[5];
    const float* ln_b   = (const float*)d_in[6];
    const float* feat   = (const float*)d_in[7];
    const float* wfeat  = (const float*)d_in[8];
    const float* bias_f = (const float*)d_in[9];
    const float* lam    = (const float*)d_in[10];
    const float* kW     = (const float*)d_in[11];   // (768, 256): K0|K1|K2
    const float* kb     = (const float*)d_in[12];
    const float* enc_g  = (const float*)d_in[13];
    const float* enc_b  = (const float*)d_in[14];
    const int*   eidx   = (const int*)d_in[15];

    const int M = in_sizes[0] / DD;       // 50000 (multiple of 16)
    const int E = in_sizes[15] / 2;       // 800000
    const int* erow = eidx;               // edge_index[0]
    const int* ecol = eidx + E;           // edge_index[1]

    const size_t MAT = (size_t)DD * DD;   // 65536

    float* ws    = (float*)d_ws;
    float* dinv  = ws;                                     // M floats (deg -> rsqrt in place)
    float* zbuf  = dinv + (((size_t)M + 3) & ~(size_t)3);  // 2*65536 (raw z, pack source)
    float* cWp   = zbuf + 2 * MAT;                         // packed conv_W  (3 matrices)
    float* lWp   = cWp + 3 * MAT;                          // packed lin_W   (3 matrices)
    float* kWp   = lWp + 3 * MAT;                          // packed kernel_W(3 matrices)
    float* zp    = kWp + 3 * MAT;                          // packed z       (2 matrices)
    float* hA    = zp + 2 * MAT;                           // current h
    float* B1    = hA + (size_t)M * DD;                    // Hw / hz scratch
    float* B2    = B1 + (size_t)M * DD;                    // agg / h2 / out_pre

    // degrees -> dinv
    k_fill1<<<(M + 255) / 256, 256, 0, stream>>>(dinv, M);
    k_deg_count<<<(E + 255) / 256, 256, 0, stream>>>(ecol, dinv, E);
    k_rsqrt_inplace<<<(M + 255) / 256, 256, 0, stream>>>(dinv, M);

    // z matrices, then pack all weight matrices into fragment-native layout
    k_compute_z<<<(2 * DD * DD) / 256, 256, 0, stream>>>(feat, wfeat, bias_f, lam, zbuf);
    k_pack<<<(3 * (int)MAT) / 256, 256, 0, stream>>>(conv_W, cWp, 3);
    k_pack<<<(3 * (int)MAT) / 256, 256, 0, stream>>>(lin_W, lWp, 3);
    k_pack<<<(3 * (int)MAT) / 256, 256, 0, stream>>>(kW, kWp, 3);
    k_pack<<<(2 * (int)MAT) / 256, 256, 0, stream>>>(zbuf, zp, 2);

    const int gblk = M / 16;
    const float* hsrc = x;
    for (int i = 0; i < 3; ++i) {
        // Hw = h @ conv_W[i]
        k_gemm<false, false><<<gblk, 128, 0, stream>>>(hsrc, cWp + i * MAT, nullptr, nullptr, B1);
        // agg = self-loop + conv_b, then edge scatter
        k_agg_init<<<(M * (DD / 4) + 255) / 256, 256, 0, stream>>>(B1, dinv, conv_b + i * DD, B2, M);
        k_scatter<<<E, DD, 0, stream>>>(B1, erow, ecol, dinv, B2);
        // h2 = agg + h @ lin_W[i] + lin_b[i]   (in-place accumulate into B2)
        k_gemm<true, true><<<gblk, 128, 0, stream>>>(hsrc, lWp + i * MAT, B2, lin_b + i * DD, B2);
        // h = relu(LN(h2))
        k_ln<true><<<(M + 7) / 8, 256, 0, stream>>>(B2, ln_g + i * DD, ln_b + i * DD, hA, M);
        hsrc = hA;
    }

    // out_pre = (h@z0)@K0 + (h@z1)@K1 + h@K2 + kernel_b
    k_gemm<false, false><<<gblk, 128, 0, stream>>>(hA, zp, nullptr, nullptr, B1);
    k_gemm<false, true ><<<gblk, 128, 0, stream>>>(B1, kWp, nullptr, kb, B2);
    k_gemm<false, false><<<gblk, 128, 0, stream>>>(hA, zp + MAT, nullptr, nullptr, B1);
    k_gemm<true,  false><<<gblk, 128, 0, stream>>>(B1, kWp + MAT, B2, nullptr, B2);
    k_gemm<true,  false><<<gblk, 128, 0, stream>>>(hA, kWp + 2 * MAT, B2, nullptr, B2);

    // final layernorm -> d_out
    k_ln<false><<<(M + 7) / 8, 256, 0, stream>>>(B2, enc_g, enc_b, (float*)d_out, M);
}